// Model_8297876816110
// MI455X (gfx1250) — compile-verified
//
#include <hip/hip_runtime.h>

// CDNA5 / gfx1250: wave32, V_WMMA_F32_16X16X4_F32 path.
// D(16x16) = A(16x4) x B(4x16) + C, chained 32x over K=128.
// One wave computes dot(e1[r], e2[r]) for 16 rows as the diagonal of
// E1 * E2^T; memory traffic is identical to a scalar dot kernel (each
// gathered W element loaded exactly once), and the workload is bandwidth
// bound (~17 MB, ~4 MFLOP), so the wasted off-diagonal FLOPs are free.

typedef __attribute__((ext_vector_type(2))) float v2f;
typedef __attribute__((ext_vector_type(8))) float v8f;

#define EMBED_DIM 128
#define BATCH     16384
#define WAVES_PER_BLOCK 8
#define ROWS_PER_WAVE   16

__global__ __launch_bounds__(WAVES_PER_BLOCK * 32)
void Model_8297876816110_kernel(const int* __restrict__ i1,
                                const int* __restrict__ i2,
                                const float* __restrict__ W1,
                                const float* __restrict__ W2,
                                const float* __restrict__ b1,
                                const float* __restrict__ b2,
                                float* __restrict__ out)
{
    const int lane = threadIdx.x & 31;
    const int wave = threadIdx.x >> 5;
    const int r0   = (blockIdx.x * WAVES_PER_BLOCK + wave) * ROWS_PER_WAVE;

    // A-matrix 16x4 f32 layout (ISA 7.12.2): lane (hi*16+m) holds
    // A[m][2*hi] in VGPR0 and A[m][2*hi+1] in VGPR1.
    // B-matrix 4x16 layout mirrors it: lane (hi*16+n) holds B[2*hi][n], B[2*hi+1][n].
    // With B = E2^T, both operands are float2 loads at (rowptr + k + 2*hi).
    const int m  = lane & 15;   // tile row (A) / tile col == row id (B)
    const int hi = lane >> 4;   // selects K sub-pair {0,1} vs {2,3}

    const int rowA = i1[r0 + m];
    const int rowB = i2[r0 + m];

    const float* pA = W1 + (size_t)rowA * EMBED_DIM + 2 * hi;
    const float* pB = W2 + (size_t)rowB * EMBED_DIM + 2 * hi;

    v8f c = {};
#pragma unroll
    for (int k = 0; k < EMBED_DIM; k += 4) {
        v2f a = *(const v2f*)(pA + k);   // global_load_b64, 8B aligned
        v2f b = *(const v2f*)(pB + k);   // global_load_b64, 8B aligned
        // 8 args: (neg_a, A, neg_b, B, c_mod, C, reuse_a, reuse_b)
        c = __builtin_amdgcn_wmma_f32_16x16x4_f32(
            /*neg_a=*/false, a, /*neg_b=*/false, b,
            /*c_mod=*/(short)0, c, /*reuse_a=*/false, /*reuse_b=*/false);
    }

    // Diagonal extraction. C/D f32 16x16 layout: VGPR v, lanes 0-15 -> M=v,
    // lanes 16-31 -> M=v+8, N = lane%16. Diag D[m][m]:
    //   lanes 0..7  : c[lane]      (m = lane)
    //   lanes 24..31: c[lane-24]   (m = lane-16)
    const int sel = lane & 7;
    float d = c[0];
#pragma unroll
    for (int v = 1; v < 8; ++v) d = (sel == v) ? c[v] : d;

    const bool writer = (lane < 8) || (lane >= 24);
    if (writer) {
        const int r = r0 + ((lane < 8) ? lane : (lane - 16));
        // rowA/rowB already equal i1[r]/i2[r] for writer lanes.
        out[r] = d + b1[rowA] + b2[rowB];
    }
}

extern "C" void kernel_launch(void* const* d_in, const int* in_sizes, int n_in,
                              void* d_out, int out_size, void* d_ws, size_t ws_size,
                              hipStream_t stream) {
    (void)in_sizes; (void)n_in; (void)d_ws; (void)ws_size; (void)out_size;
    const int*   i1 = (const int*)  d_in[0];
    const int*   i2 = (const int*)  d_in[1];
    const float* W1 = (const float*)d_in[2];
    const float* W2 = (const float*)d_in[3];
    const float* b1 = (const float*)d_in[4];
    const float* b2 = (const float*)d_in[5];
    float* out = (float*)d_out;

    const int rows_per_block = WAVES_PER_BLOCK * ROWS_PER_WAVE;   // 128
    dim3 grid(BATCH / rows_per_block);                            // 128 blocks
    dim3 block(WAVES_PER_BLOCK * 32);                             // 256 threads
    Model_8297876816110_kernel<<<grid, block, 0, stream>>>(i1, i2, W1, W2, b1, b2, out);
}